// GATLayer_81587198755202
// MI455X (gfx1250) — compile-verified
//
#include <hip/hip_runtime.h>
#include <cstdint>

typedef __attribute__((ext_vector_type(2))) float v2f;
typedef __attribute__((ext_vector_type(8))) float v8f;

#define NN 8192
#define DIN 256
#define DOUT 64

__device__ __forceinline__ v8f wmma_f32_k4(v2f a, v2f b, v8f c) {
    // V_WMMA_F32_16X16X4_F32: D = A(16x4,f32) x B(4x16,f32) + C(16x16,f32)
    return __builtin_amdgcn_wmma_f32_16x16x4_f32(
        /*neg_a=*/false, a, /*neg_b=*/false, b,
        /*c_mod=*/(short)0, c, /*reuse_a=*/false, /*reuse_b=*/false);
}

// ---------------------------------------------------------------------------
// Kernel 1: Wh = h @ w   [8192x256]@[256x64] -> [8192x64], f32 WMMA 16x16x4.
// 128 threads = 4 waves; wave wv computes row-tile (blockIdx.x*4+wv): 16x64.
// ---------------------------------------------------------------------------
__global__ void k_gemm_wh(const float* __restrict__ h,
                          const float* __restrict__ w,
                          float* __restrict__ Wh) {
    const int lane = threadIdx.x & 31;
    const int wv   = threadIdx.x >> 5;
    const int hi   = lane >> 4;       // 0: lanes 0-15, 1: lanes 16-31
    const int lm   = lane & 15;
    const int rowTile = blockIdx.x * 4 + wv;          // 0..511
    const int arow = rowTile * 16 + lm;               // A-matrix row for this lane
    const float* hrow = h + (size_t)arow * DIN;

    v8f acc[4] = {};
    for (int k = 0; k < DIN; k += 4) {
        const float2 av = *(const float2*)(hrow + k + 2 * hi);  // K=2*hi, 2*hi+1
        v2f a; a.x = av.x; a.y = av.y;
#pragma unroll
        for (int nt = 0; nt < 4; ++nt) {
            const int col = nt * 16 + lm;
            v2f b;
            b.x = w[(size_t)(k + 2 * hi + 0) * DOUT + col];  // B: vgpr0 = row K=2*hi
            b.y = w[(size_t)(k + 2 * hi + 1) * DOUT + col];  // B: vgpr1 = row K=2*hi+1
            acc[nt] = wmma_f32_k4(a, b, acc[nt]);
        }
    }
    // C layout: vgpr v -> row v + 8*hi ; col = lm + 16*nt
#pragma unroll
    for (int nt = 0; nt < 4; ++nt)
#pragma unroll
        for (int v = 0; v < 8; ++v)
            Wh[(size_t)(rowTile * 16 + v + 8 * hi) * DOUT + nt * 16 + lm] = acc[nt][v];
}

// ---------------------------------------------------------------------------
// Kernel: init max-key to -inf encoding (d_ws is poisoned, must self-init).
// ---------------------------------------------------------------------------
__global__ void k_init(int* maxkey) { *maxkey = INT32_MIN; }

// ---------------------------------------------------------------------------
// Kernel 2: Wh1[i] = Wh[i]·a[:64], Wh2[i] = Wh[i]·a[64:], atomicMax of Wh2
// via order-preserving signed-int key: key = bits>=0 ? bits : bits^0x7FFFFFFF
// ---------------------------------------------------------------------------
__global__ void k_rowproj(const float* __restrict__ Wh,
                          const float* __restrict__ a,
                          float* __restrict__ Wh1,
                          float* __restrict__ Wh2,
                          int* __restrict__ maxkey) {
    const int i = blockIdx.x * 256 + threadIdx.x;
    const float* r = Wh + (size_t)i * DOUT;
    float s1 = 0.f, s2 = 0.f;
#pragma unroll 8
    for (int c = 0; c < DOUT; ++c) {
        const float x = r[c];
        s1 += x * a[c];
        s2 += x * a[DOUT + c];
    }
    Wh1[i] = s1;
    Wh2[i] = s2;
    const int bits = __float_as_int(s2);
    const int key = bits >= 0 ? bits : (bits ^ 0x7FFFFFFF);
    atomicMax(maxkey, key);
}

// ---------------------------------------------------------------------------
// Kernel 3: fused flash-GAT.
// Block = 256 thr (8 waves) owns a 16-row tile. Wave wv streams j-chunks of 4
// (stride 32). Per chunk: p = adj>0 ? exp(leakyrelu(Wh1[i]+Wh2[j]) - m_i) : 0,
// then acc[16x64] += p(16x4) @ Wh(4x64) via 4x WMMA f32 16x16x4.
// m_i = leakyrelu(Wh1[i] + max_j Wh2[j]) >= true row max (monotone) => partial
// accumulators across waves add directly; single LDS combine, normalize, relu.
// adj & Wh2 fetched as 8B vector loads (c0 = j+2*hi is even, bases 8B-aligned).
// ---------------------------------------------------------------------------
__global__ void k_gat_flash(const float* __restrict__ Wh,
                            const float* __restrict__ Wh1v,
                            const float* __restrict__ Wh2v,
                            const int* __restrict__ adj,
                            const int* __restrict__ maxkey,
                            float* __restrict__ out) {
    __shared__ float sAcc[8][1024];   // [wave][nt*256 + v*32 + lane]  (32 KB)
    __shared__ float sS[8][32];       // per-wave, per-lane row-sum partials

    const int tid  = threadIdx.x;
    const int wv   = tid >> 5;
    const int lane = tid & 31;
    const int hi   = lane >> 4;
    const int lm   = lane & 15;
    const int rowTile = blockIdx.x;                   // 0..511
    const int arow = rowTile * 16 + lm;               // score row for this lane

    const float wh1 = Wh1v[arow];
    const int key = *maxkey;
    const int mb  = key >= 0 ? key : (key ^ 0x7FFFFFFF);
    const float maxWh2 = __int_as_float(mb);
    float t = wh1 + maxWh2;
    const float mrow = t > 0.f ? t : 0.2f * t;        // per-row softmax shift (upper bound)

    const int2*   adjrow2 = (const int2*)(adj + (size_t)arow * NN);
    const float2* wh2p    = (const float2*)Wh2v;

    v8f acc[4] = {};
    float srow = 0.f;

    for (int j = wv * 4; j < NN; j += 32) {
        const int c0   = j + 2 * hi;                  // this lane's two K columns
        const int half = c0 >> 1;
        const int2   m  = adjrow2[half];              // adj[arow][c0], adj[arow][c0+1]
        const float2 w2 = wh2p[half];                 // Wh2[c0], Wh2[c0+1]

        float e0 = wh1 + w2.x;
        float e1 = wh1 + w2.y;
        e0 = e0 > 0.f ? e0 : 0.2f * e0;
        e1 = e1 > 0.f ? e1 : 0.2f * e1;
        const float p0 = m.x > 0 ? __expf(e0 - mrow) : 0.f;
        const float p1 = m.y > 0 ? __expf(e1 - mrow) : 0.f;
        srow += p0 + p1;

        v2f pa; pa.x = p0; pa.y = p1;                 // A frag: K=2*hi, 2*hi+1
        const float* whr = Wh + (size_t)c0 * DOUT;    // B rows c0, c0+1
#pragma unroll
        for (int nt = 0; nt < 4; ++nt) {
            const int col = nt * 16 + lm;
            v2f b;
            b.x = whr[col];
            b.y = whr[DOUT + col];
            acc[nt] = wmma_f32_k4(pa, b, acc[nt]);
        }
    }

    // Dump partials to LDS.
#pragma unroll
    for (int nt = 0; nt < 4; ++nt)
#pragma unroll
        for (int v = 0; v < 8; ++v)
            sAcc[wv][nt * 256 + v * 32 + lane] = acc[nt][v];
    sS[wv][lane] = srow;
    __syncthreads();

    // Combine 8 waves, normalize, relu, store. 1024 outputs / 256 threads.
    for (int e = tid; e < 16 * DOUT; e += 256) {
        const int r  = e >> 6;          // 0..15
        const int c  = e & 63;
        const int v  = r & 7;
        const int h2 = r >> 3;
        const int ln = h2 * 16 + (c & 15);
        const int nt = c >> 4;
        float num = 0.f, den = 0.f;
#pragma unroll
        for (int w8 = 0; w8 < 8; ++w8) {
            num += sAcc[w8][nt * 256 + v * 32 + ln];
            den += sS[w8][r] + sS[w8][r + 16];        // both half-wave partials
        }
        float o = den > 0.f ? num / den : 0.f;
        out[(size_t)(rowTile * 16 + r) * DOUT + c] = o > 0.f ? o : 0.f;
    }
}

// ---------------------------------------------------------------------------
extern "C" void kernel_launch(void* const* d_in, const int* in_sizes, int n_in,
                              void* d_out, int out_size, void* d_ws, size_t ws_size,
                              hipStream_t stream) {
    const float* h   = (const float*)d_in[0];   // [8192,256]
    const int*   adj = (const int*)  d_in[1];   // [8192,8192]
    const float* w   = (const float*)d_in[2];   // [256,64]
    const float* a   = (const float*)d_in[3];   // [128,1]
    float* out = (float*)d_out;                 // [8192,64]

    // Workspace layout: Wh (2 MB) | Wh1 (32 KB) | Wh2 (32 KB) | maxkey (4 B)
    float* Wh   = (float*)d_ws;
    float* Wh1  = Wh  + (size_t)NN * DOUT;
    float* Wh2  = Wh1 + NN;
    int*   mkey = (int*)(Wh2 + NN);

    k_gemm_wh<<<NN / 16 / 4, 128, 0, stream>>>(h, w, Wh);
    k_init<<<1, 1, 0, stream>>>(mkey);
    k_rowproj<<<NN / 256, 256, 0, stream>>>(Wh, a, Wh1, Wh2, mkey);
    k_gat_flash<<<NN / 16, 256, 0, stream>>>(Wh, Wh1, Wh2, adj, mkey, out);
}